// ScaledDotProductAttention_57836029608220
// MI455X (gfx1250) — compile-verified
//
#include <hip/hip_runtime.h>

typedef __attribute__((ext_vector_type(2))) float v2f;
typedef __attribute__((ext_vector_type(8))) float v8f;

#define S_LEN   4096
#define DK      16
#define LSTRIDE 4132          // 4096 + 36 words: row pad for LDS bank-conflict freedom
#define NWAVES  8
#define KTILES  (S_LEN / 16)  // 256

// Dynamic LDS layout (floats):
//   [0, 16*LSTRIDE)             score / P slab, 16 rows x 4096 cols (padded)
//   [16*LSTRIDE, +256)          per-thread reduction partials
//   [+256, +16)                 row max
//   [+16,  +16)                 row 1/sum
#define SMEM_FLOATS (16 * LSTRIDE + 256 + 16 + 16)

__global__ __launch_bounds__(256)
void attn_sdpa_wmma_kernel(const float* __restrict__ Q,
                           const float* __restrict__ Kmat,
                           const float* __restrict__ V,
                           const unsigned char* __restrict__ mask, // jnp.bool_ -> 1 byte
                           const float* __restrict__ link,
                           const float* __restrict__ sw,
                           const float* __restrict__ sb,
                           float* __restrict__ ctx_out,
                           float* __restrict__ attn_out)
{
    extern __shared__ float smem[];
    float* sc     = smem;                   // [16][LSTRIDE]
    float* red    = smem + 16 * LSTRIDE;    // [256]
    float* rowmax = red + 256;              // [16]
    float* rowinv = rowmax + 16;            // [16]

    const int tid  = threadIdx.x;
    const int wave = tid >> 5;
    const int lane = tid & 31;
    const int m    = lane & 15;   // row (A) / col (B,C) within 16
    const int h    = lane >> 4;   // lane-half selects K/M offset per ISA layout
    const int qt   = blockIdx.x;  // q tile 0..255
    const int b    = blockIdx.y;  // batch 0..3

    const int   qbase  = b * S_LEN + qt * 16;       // global row index base
    const float simw   = sw[0];
    const float simb   = sb[0];
    const float scale  = 0.25f;                     // 1/sqrt(16)

    // ---------------- Q tile in WMMA f32 16x16x4 A-layout ----------------
    // VGPR0: K=4j+2h, VGPR1: K=4j+2h+1 for lane-half h, row M=m.
    v2f a[4];
#pragma unroll
    for (int j = 0; j < 4; ++j) {
        a[j] = *(const v2f*)(Q + (size_t)(qbase + m) * DK + 4 * j + 2 * h);
    }

    // ---------------- Phase A: score tiles via WMMA ----------------
    // All loads of an iteration are issued before any consumer so the
    // compiler can overlap the link/mask gathers with the WMMA chain.
    for (int kt = wave; kt < KTILES; kt += NWAVES) {
        const int kcol0 = kt * 16;

        // 1) K tile (B-matrix) loads — issued first; WMMA waits only on these
        v2f bm[4];
#pragma unroll
        for (int j = 0; j < 4; ++j)
            bm[j] = *(const v2f*)(Kmat + ((size_t)b * S_LEN + kcol0 + m) * DK + 4 * j + 2 * h);

        // 2) link / mask gathers — in flight while the WMMAs execute
        float lk[8];
        unsigned char mk[8];
#pragma unroll
        for (int i = 0; i < 8; ++i) {
            const size_t gi = (size_t)(qbase + i + 8 * h) * S_LEN + kcol0 + m;
            lk[i] = link[gi];
            mk[i] = mask[gi];
        }

        // 3) prefetch next iteration's link rows (largest read stream) into L2
        if (kt + NWAVES < KTILES) {
#pragma unroll
            for (int i = 0; i < 8; ++i) {
                const size_t gn = (size_t)(qbase + i + 8 * h) * S_LEN
                                + (kcol0 + NWAVES * 16) + m;
                __builtin_prefetch(&link[gn], 0, 1);   // global_prefetch_b8
            }
        }

        // 4) QK^T for this 16x16 tile: 4 chained f32 WMMAs over K=4 each
        v8f c = {};
#pragma unroll
        for (int j = 0; j < 4; ++j)
            c = __builtin_amdgcn_wmma_f32_16x16x4_f32(false, a[j], false, bm[j],
                                                      (short)0, c, false, false);

        // 5) epilogue: scale + scalar 1x1-conv affine + mask, store to LDS slab
        //    C/D layout: c[i] = scores[row = i + 8h][col = kcol0 + m]
#pragma unroll
        for (int i = 0; i < 8; ++i) {
            float s = c[i] * scale + simw * lk[i] + simb;
            s = mk[i] ? -1.0e9f : s;
            sc[(i + 8 * h) * LSTRIDE + kcol0 + m] = s;
        }
    }
    __syncthreads();

    // ---------------- Phase B: softmax over each of the 16 rows ----------------
    const int r  = tid >> 4;   // row 0..15
    const int cc = tid & 15;   // 16 threads cooperate per row, stride-16 columns
    float mx = -__builtin_inff();
    for (int i = 0; i < 256; ++i)
        mx = fmaxf(mx, sc[r * LSTRIDE + cc + 16 * i]);
    red[tid] = mx;
    __syncthreads();
    if (tid < 16) {
        float mm = red[tid * 16];
#pragma unroll
        for (int i = 1; i < 16; ++i) mm = fmaxf(mm, red[tid * 16 + i]);
        rowmax[tid] = mm;
    }
    __syncthreads();
    const float rm = rowmax[r];
    float sum = 0.f;
    for (int i = 0; i < 256; ++i) {
        const int idx = r * LSTRIDE + cc + 16 * i;
        const float e = __expf(sc[idx] - rm);
        sc[idx] = e;                      // cache exp for normalize pass
        sum += e;
    }
    red[tid] = sum;
    __syncthreads();
    if (tid < 16) {
        float ss = 0.f;
#pragma unroll
        for (int i = 0; i < 16; ++i) ss += red[tid * 16 + i];
        rowinv[tid] = 1.0f / ss;
    }
    __syncthreads();

    // normalize in LDS + coalesced global store of attn
    for (int idx = tid; idx < 16 * S_LEN; idx += 256) {
        const int row = idx >> 12;
        const int col = idx & (S_LEN - 1);
        const float p = sc[row * LSTRIDE + col] * rowinv[row];
        sc[row * LSTRIDE + col] = p;
        attn_out[(size_t)(qbase + row) * S_LEN + col] = p;
    }
    __syncthreads();

    // ---------------- Phase C: context = P x V via WMMA ----------------
    v8f acc = {};
    for (int kt = wave; kt < KTILES; kt += NWAVES) {
        const int kcol0 = kt * 16;

        // V (B-matrix) gathers — 8 scalar global loads issued up front
        float vb[8];
#pragma unroll
        for (int j = 0; j < 4; ++j) {
            const float* vp = V + ((size_t)b * S_LEN + kcol0 + 4 * j + 2 * h) * DK + m;
            vb[2 * j]     = vp[0];
            vb[2 * j + 1] = vp[DK];
        }
        // P (A-matrix) from LDS — 4 conflict-free ds_load_b64, independent counter
        v2f ap[4];
#pragma unroll
        for (int j = 0; j < 4; ++j)
            ap[j] = *(const v2f*)&sc[m * LSTRIDE + kcol0 + 4 * j + 2 * h];

#pragma unroll
        for (int j = 0; j < 4; ++j) {
            v2f bp;
            bp[0] = vb[2 * j];
            bp[1] = vb[2 * j + 1];
            acc = __builtin_amdgcn_wmma_f32_16x16x4_f32(false, ap[j], false, bp,
                                                        (short)0, acc, false, false);
        }
    }
    __syncthreads();            // all LDS reads of P done before slab reuse

    // cross-wave reduction of 16x16 partials through LDS
#pragma unroll
    for (int i = 0; i < 8; ++i)
        smem[wave * 256 + (i + 8 * h) * 16 + m] = acc[i];
    __syncthreads();
    {
        float s2 = 0.f;
#pragma unroll
        for (int w = 0; w < NWAVES; ++w) s2 += smem[w * 256 + tid];
        ctx_out[(size_t)(qbase + (tid >> 4)) * DK + (tid & 15)] = s2;
    }
}

extern "C" void kernel_launch(void* const* d_in, const int* in_sizes, int n_in,
                              void* d_out, int out_size, void* d_ws, size_t ws_size,
                              hipStream_t stream)
{
    (void)in_sizes; (void)n_in; (void)out_size; (void)d_ws; (void)ws_size;

    const float*         Q    = (const float*)d_in[0];
    const float*         Km   = (const float*)d_in[1];
    const float*         V    = (const float*)d_in[2];
    const unsigned char* mask = (const unsigned char*)d_in[3]; // bool -> 1 byte
    const float*         link = (const float*)d_in[4];
    const float*         sw   = (const float*)d_in[5];
    const float*         sb   = (const float*)d_in[6];

    float* ctx  = (float*)d_out;                       // [4,1,4096,16]
    float* attn = ctx + (size_t)4 * S_LEN * DK;        // [4,1,4096,4096]

    const size_t shmem = (size_t)SMEM_FLOATS * sizeof(float); // ~258 KB, fits 320 KB WGP LDS
    (void)hipFuncSetAttribute((const void*)attn_sdpa_wmma_kernel,
                              hipFuncAttributeMaxDynamicSharedMemorySize, (int)shmem);

    dim3 grid(KTILES, 4);   // 256 q-tiles x 4 batches = 1024 workgroups
    dim3 block(256);        // 8 wave32
    attn_sdpa_wmma_kernel<<<grid, block, shmem, stream>>>(Q, Km, V, mask, link,
                                                          sw, sb, ctx, attn);
}